// FocalCTCLoss_26869315403817
// MI455X (gfx1250) — compile-verified
//
#include <hip/hip_runtime.h>
#include <hip/hip_bf16.h>
#include <stdint.h>
#include <stddef.h>

#define T_DIM 1000
#define B_DIM 64
#define C_DIM 256
#define L_DIM 100
#define S_DIM (2 * L_DIM + 1)   // 201
#define BC    (B_DIM * C_DIM)   // 16384
#define NEGV  (-1e30f)

#define PF_D   7                // async prefetch depth (rows in flight)
#define NBUF   8                // ring buffers (power of two: % -> &)
#define BMSK   (NBUF - 1)

// ---------------- CDNA5 async-to-LDS path (guarded) ----------------
#if defined(__has_builtin)
#  if __has_builtin(__builtin_amdgcn_global_load_async_to_lds_b32) && \
      __has_builtin(__builtin_amdgcn_s_wait_asynccnt)
#    define USE_ASYNC_LDS 1
#  endif
#endif
#ifndef USE_ASYNC_LDS
#  define USE_ASYNC_LDS 0
#endif

__device__ __forceinline__ void row_prefetch(const float* gsrc, float* ldst) {
#if USE_ASYNC_LDS
  __builtin_amdgcn_global_load_async_to_lds_b32(
      (__attribute__((address_space(1))) int*)const_cast<float*>(gsrc),
      (__attribute__((address_space(3))) int*)ldst,
      /*imm offset*/ 0, /*cpol*/ 0);
#else
  *ldst = *gsrc;   // synchronous fallback staging into LDS
#endif
}

// ---------------- Kernel 1: mean_t exp(log_probs)  -> (B*C) ----------------
__global__ __launch_bounds__(256) void mean_exp_kernel(
    const float* __restrict__ lp, float* __restrict__ meanp) {
  const int col = blockIdx.x * 256 + threadIdx.x;   // < B*C, coalesced per t
  const float* p = lp + col;
  float acc = 0.0f;
#pragma unroll 4
  for (int t = 0; t < T_DIM; ++t)
    acc += __expf(p[(size_t)t * BC]);
  meanp[col] = acc * (1.0f / (float)T_DIM);
}

// ---------------- Kernel 2: sum of focal weights over B*L targets ----------
__global__ __launch_bounds__(256) void focal_kernel(
    const int* __restrict__ targets, const float* __restrict__ meanp,
    float* __restrict__ out_sum) {
  __shared__ float red[256];
  float acc = 0.0f;
  for (int j = threadIdx.x; j < B_DIM * L_DIM; j += 256) {
    const int b = j / L_DIM;
    const float pb = meanp[b * C_DIM + targets[j]];
    const float w = 1.0f - pb;
    acc += w * w;
  }
  red[threadIdx.x] = acc;
  __syncthreads();
  for (int o = 128; o > 0; o >>= 1) {
    if (threadIdx.x < o) red[threadIdx.x] += red[threadIdx.x + o];
    __syncthreads();
  }
  if (threadIdx.x == 0) out_sum[0] = red[0];
}

// ---------------- Kernel 3: CTC alpha recursion, one block per batch -------
__device__ __forceinline__ void capture_loss(const float* abuf, int tlen,
                                             float* slot) {
  const float x = abuf[2 + 2 * tlen];
  const float y = abuf[2 + 2 * tlen - 1];   // tlen==0 -> pad slot (NEG), safe
  const float m = fmaxf(x, y);
  const float ll = m + __logf(__expf(x - m) + __expf(y - m));
  *slot = (ll > -5e29f) ? -ll : 0.0f;       // zero_infinity semantics
}

__global__ __launch_bounds__(256) void ctc_kernel(
    const float* __restrict__ lp, const int* __restrict__ targets,
    const int* __restrict__ ilen, const int* __restrict__ tlen,
    float* __restrict__ loss_out) {
  const int b = blockIdx.x;
  const int tid = threadIdx.x;

  __shared__ float row[NBUF][C_DIM];       // ring of emission rows (8 KB)
  __shared__ float alpha[2][S_DIM + 2];    // [0..1] = NEG pads, states at [2+s]
  __shared__ int   tgt[L_DIM];

  const float* lpb = lp + (size_t)b * C_DIM;   // &lp[(t*B + b)*C], t = 0
  const int myIlen = ilen[b];
  const int myTlen = tlen[b];

  if (tid < L_DIM) tgt[tid] = targets[b * L_DIM + tid];
  row[0][tid] = lpb[tid];                      // t = 0 row (C_DIM == 256)
  __syncthreads();

  // loop-invariant per-state registers (state s == tid)
  const int  ext_r  = (tid & 1) ? tgt[tid >> 1] : 0;
  const bool allow  = (tid >= 3) && (tid & 1) &&
                      (tgt[tid >> 1] != tgt[(tid >> 1) - 1]);
  const int  a2idx  = allow ? tid : 0;         // slot 0 is a permanent NEG pad

  // alpha(0) init: LDS copy + register copy of own state
  float myA = NEGV;
  if (tid == 0) myA = row[0][0];
  else if (tid == 1) myA = row[0][tgt[0]];
  if (tid < S_DIM + 2) {
    float v = NEGV;
    if (tid == 2)      v = row[0][0];
    else if (tid == 3) v = row[0][tgt[0]];
    alpha[0][tid] = v;                         // includes pads [0..1]
  }
  if (tid < 2) alpha[1][tid] = NEGV;           // pads of ping buffer

  // preload rows 1..PF_D into the ring (ASYNCcnt-tracked, in-order retire)
#pragma unroll
  for (int tp = 1; tp <= PF_D; ++tp)
    row_prefetch(lpb + (size_t)tp * BC + tid, &row[tp & BMSK][tid]);
  __syncthreads();

  const float* gpre = lpb + (size_t)(PF_D + 1) * BC + tid;  // next row to issue
  int cur = 0;

  // ---- steady state: PF_D rows in flight, wait asynccnt <= PF_D-1 ----------
  for (int t = 1; t < T_DIM - PF_D; ++t) {
#if USE_ASYNC_LDS
    __builtin_amdgcn_s_wait_asynccnt(PF_D - 1);  // oldest (row t) has landed
#endif
    __syncthreads();

    if (tid == 0 && t == myIlen)                 // snapshot alpha(t-1)
      capture_loss(alpha[cur], myTlen, loss_out + b);

    row_prefetch(gpre, &row[(t + PF_D) & BMSK][tid]);  // == (t-1)&BMSK: safe
    gpre += BC;

    const int nxt = cur ^ 1;
    if (tid < S_DIM) {
      const float a0 = myA;
      const float a1 = alpha[cur][1 + tid];
      const float a2 = alpha[cur][a2idx];
      const float m  = fmaxf(a0, fmaxf(a1, a2));
      const float s3 = __expf(a0 - m) + __expf(a1 - m) + __expf(a2 - m);
      myA = m + __logf(s3) + row[t & BMSK][ext_r];
      alpha[nxt][2 + tid] = myA;
    }
    cur = nxt;
  }

  // ---- tail: fewer than PF_D outstanding; drain all ------------------------
  for (int t = (T_DIM - PF_D > 1 ? T_DIM - PF_D : 1); t < T_DIM; ++t) {
#if USE_ASYNC_LDS
    __builtin_amdgcn_s_wait_asynccnt(0);
#endif
    __syncthreads();

    if (tid == 0 && t == myIlen)
      capture_loss(alpha[cur], myTlen, loss_out + b);

    const int nxt = cur ^ 1;
    if (tid < S_DIM) {
      const float a0 = myA;
      const float a1 = alpha[cur][1 + tid];
      const float a2 = alpha[cur][a2idx];
      const float m  = fmaxf(a0, fmaxf(a1, a2));
      const float s3 = __expf(a0 - m) + __expf(a1 - m) + __expf(a2 - m);
      myA = m + __logf(s3) + row[t & BMSK][ext_r];
      alpha[nxt][2 + tid] = myA;
    }
    cur = nxt;
  }

  __syncthreads();
  if (tid == 0 && myIlen == T_DIM)
    capture_loss(alpha[cur], myTlen, loss_out + b);
}

// ---------------- Kernel 4: final scalar combine ---------------------------
__global__ __launch_bounds__(64) void combine_kernel(
    const float* __restrict__ lossv, const float* __restrict__ fsum,
    float* __restrict__ out) {
  __shared__ float red[64];
  red[threadIdx.x] = lossv[threadIdx.x];
  __syncthreads();
  for (int o = 32; o > 0; o >>= 1) {
    if (threadIdx.x < o) red[threadIdx.x] += red[threadIdx.x + o];
    __syncthreads();
  }
  if (threadIdx.x == 0)
    out[0] = red[0] * fsum[0] *
             (1.0f / ((float)B_DIM * (float)(B_DIM * L_DIM)));
}

// ---------------- Host entry ----------------------------------------------
extern "C" void kernel_launch(void* const* d_in, const int* in_sizes, int n_in,
                              void* d_out, int out_size, void* d_ws,
                              size_t ws_size, hipStream_t stream) {
  (void)in_sizes; (void)n_in; (void)out_size; (void)ws_size;
  const float* lp      = (const float*)d_in[0];  // (T,B,C) f32
  const int*   targets = (const int*)d_in[1];    // (B*L,)  i32
  const int*   ilen    = (const int*)d_in[2];    // (B,)    i32
  const int*   tlen    = (const int*)d_in[3];    // (B,)    i32

  float* ws    = (float*)d_ws;
  float* meanp = ws;             // BC floats
  float* fsum  = ws + BC;        // 1 float
  float* lossv = ws + BC + 16;   // B floats (padded offset)

  mean_exp_kernel<<<BC / 256, 256, 0, stream>>>(lp, meanp);
  focal_kernel<<<1, 256, 0, stream>>>(targets, meanp, fsum);
  ctc_kernel<<<B_DIM, 256, 0, stream>>>(lp, targets, ilen, tlen, lossv);
  combine_kernel<<<1, 64, 0, stream>>>(lossv, fsum, (float*)d_out);
}